// GAT_52501680226730
// MI455X (gfx1250) — compile-verified
//
#include <hip/hip_runtime.h>
#include <stdint.h>

typedef __attribute__((ext_vector_type(2))) float v2f;
typedef __attribute__((ext_vector_type(8))) float v8f;

#define HID 64
#define NEG_SLOPE 0.2f
#define KSTEP 32   // K-chunk staged in LDS per iteration

// ---------------------------------------------------------------------------
// C[M x Nd] = A[M x K] * B[K x Nd], f32 row-major, via V_WMMA_F32_16X16X4_F32.
//
// Block = 128 threads (4 waves). Block tile = 16 M rows x 64 N cols:
//   - the 16xKSTEP A tile is staged once into LDS with
//     GLOBAL_LOAD_ASYNC_TO_LDS_B128 (one 16B async transfer per thread,
//     ASYNCcnt-tracked), then shared by all 4 waves -> 4x less A traffic.
//   - wave w computes the 16x16 N-tile at n0 = blockIdx.y*64 + w*16.
// Fragment layouts (ISA 7.12.2, 32-bit operands):
//   A-frag 16x4: lane L -> (m = L&15, k = 2*(L>>4)) and k+1   (2 VGPRs)
//   B-frag 4x16: lane L -> (k = 2*(L>>4), n = L&15) and k+1   (2 VGPRs)
//   C/D 16x16 :  vgpr i on lane L -> row (i + 8*(L>>4)), col (L&15)
// Requires: M % 16 == 0, Nd % 64 == 0 or == 64*grid.y, K % KSTEP == 0.
// ---------------------------------------------------------------------------
__global__ __launch_bounds__(128) void gemm_wmma_f32(const float* __restrict__ A,
                                                     const float* __restrict__ B,
                                                     float* __restrict__ C,
                                                     int M, int K, int Nd) {
  __shared__ float sA[16 * KSTEP];  // 2 KB

  const int tid = threadIdx.x;
  const int lane = tid & 31;
  const int wid = tid >> 5;           // 0..3
  const int mn = lane & 15;
  const int half = lane >> 4;

  const int m0 = blockIdx.x * 16;
  const int n0 = blockIdx.y * 64 + wid * 16;

  // staging assignment: thread t copies A[m0 + (t>>3)][k0 + (t&7)*4 .. +3]
  const int srow = tid >> 3;
  const int scol = (tid & 7) * 4;
  const float* __restrict__ astage = A + (size_t)(m0 + srow) * K + scol;
  const uint32_t lds_off = (uint32_t)(uintptr_t)&sA[srow * KSTEP + scol];

  const float* __restrict__ bcol = B + (size_t)n0 + mn;

  v8f acc = {};
  for (int k0 = 0; k0 < K; k0 += KSTEP) {
    // ---- async stage A tile: global -> LDS, 16 bytes per thread ----------
    {
      const float* gaddr = astage + k0;
      asm volatile("global_load_async_to_lds_b128 %0, %1, off"
                   :: "v"(lds_off), "v"(gaddr)
                   : "memory");
      asm volatile("s_wait_asynccnt 0x0" ::: "memory");
    }
    __syncthreads();

    // ---- 8 WMMA steps over the staged K-chunk ----------------------------
#pragma unroll
    for (int kk = 0; kk < KSTEP; kk += 4) {
      const int ka = kk + 2 * half;
      float2 av = *reinterpret_cast<const float2*>(&sA[mn * KSTEP + ka]);
      v2f a, b;
      a.x = av.x;
      a.y = av.y;
      b.x = bcol[(size_t)(k0 + ka) * Nd];
      b.y = bcol[(size_t)(k0 + ka + 1) * Nd];
      acc = __builtin_amdgcn_wmma_f32_16x16x4_f32(false, a, false, b,
                                                  (short)0, acc, false, false);
    }
    __syncthreads();  // all waves done reading sA before next stage
  }

  float* __restrict__ crow = C + (size_t)(m0 + 8 * half) * Nd + n0 + mn;
#pragma unroll
  for (int i = 0; i < 8; ++i) crow[(size_t)i * Nd] = acc[i];
}

// ---------------------------------------------------------------------------
// Per (node, head): el = <feat, al>, er = <feat, ar>; also reset emax/denom.
// feat layout [N, H, HID] flattened; gid = n*H + h.
// ---------------------------------------------------------------------------
__global__ void attn_prep(const float* __restrict__ feat,
                          const float* __restrict__ al,
                          const float* __restrict__ ar,
                          float* __restrict__ el, float* __restrict__ er,
                          float* __restrict__ emax, float* __restrict__ denom,
                          int NH, int H) {
  int gid = blockIdx.x * blockDim.x + threadIdx.x;
  if (gid >= NH) return;
  int h = gid % H;
  const float* __restrict__ f = feat + (size_t)gid * HID;
  const float* __restrict__ av = al + h * HID;
  const float* __restrict__ rv = ar + h * HID;
  float sl = 0.f, sr = 0.f;
#pragma unroll 4
  for (int k = 0; k < HID; ++k) {
    float v = f[k];
    sl += v * av[k];
    sr += v * rv[k];
  }
  el[gid] = sl;
  er[gid] = sr;
  emax[gid] = -3.0e38f;
  denom[gid] = 0.f;
}

// Out accumulator starts at the bias, scatter adds on top -> final result.
__global__ void init_bias_k(float* __restrict__ out, const float* __restrict__ b,
                            int total, int HF) {
  int gid = blockIdx.x * blockDim.x + threadIdx.x;
  if (gid < total) out[gid] = b[gid % HF];
}

__device__ __forceinline__ void edge_sd(const int* __restrict__ src,
                                        const int* __restrict__ dst,
                                        int E, int idx, int& s, int& d) {
  if (idx < E) {
    s = src[idx];
    d = dst[idx];
  } else {            // implicit self loop
    s = idx - E;
    d = idx - E;
  }
}

__device__ __forceinline__ float lrelu(float x) {
  return x > 0.f ? x : NEG_SLOPE * x;
}

// Pass 1: segment max over destinations (stable softmax).
__global__ void edge_max_k(const int* __restrict__ src, const int* __restrict__ dst,
                           int E, int EtotH,
                           const float* __restrict__ el, const float* __restrict__ er,
                           float* __restrict__ emax, int H) {
  int gid = blockIdx.x * blockDim.x + threadIdx.x;
  if (gid >= EtotH) return;
  int idx = gid / H, h = gid - idx * H;
  int s, d;
  edge_sd(src, dst, E, idx, s, d);
  float e = lrelu(el[s * H + h] + er[d * H + h]);
  atomicMax(&emax[d * H + h], e);
}

// Pass 2: ex = exp(e - max); cache ex per edge; segment-sum denominators.
__global__ void edge_expsum_k(const int* __restrict__ src, const int* __restrict__ dst,
                              int E, int EtotH,
                              const float* __restrict__ el, const float* __restrict__ er,
                              const float* __restrict__ emax,
                              float* __restrict__ exbuf, float* __restrict__ denom, int H) {
  int gid = blockIdx.x * blockDim.x + threadIdx.x;
  if (gid >= EtotH) return;
  int idx = gid / H, h = gid - idx * H;
  int s, d;
  edge_sd(src, dst, E, idx, s, d);
  float e = lrelu(el[s * H + h] + er[d * H + h]);
  float ex = __expf(e - emax[d * H + h]);
  exbuf[gid] = ex;
  atomicAdd(&denom[d * H + h], ex);
}

// Pass 3: weighted scatter, one wave per edge; 32 lanes stride the H*HID feats.
__global__ __launch_bounds__(256) void edge_scatter_k(const int* __restrict__ src,
                                                      const int* __restrict__ dst,
                                                      int E, int Etot,
                                                      const float* __restrict__ feat,
                                                      const float* __restrict__ exbuf,
                                                      const float* __restrict__ denom,
                                                      float* __restrict__ out, int H) {
  int widx = blockIdx.x * 8 + (threadIdx.x >> 5);
  if (widx >= Etot) return;
  int lane = threadIdx.x & 31;
  int s, d;
  edge_sd(src, dst, E, widx, s, d);
  const int HF = H * HID;
  const float* __restrict__ frow = feat + (size_t)s * HF;
  float* __restrict__ orow = out + (size_t)d * HF;
  for (int c = lane; c < HF; c += 32) {
    int h = c >> 6;  // HID == 64
    float alpha = exbuf[widx * H + h] / denom[d * H + h];
    atomicAdd(&orow[c], frow[c] * alpha);
  }
}

// ---------------------------------------------------------------------------
// One GAT layer: gemm -> attn coeffs -> bias-init -> 3 edge passes.
// ---------------------------------------------------------------------------
static void run_layer(const float* h, const float* W, const float* al, const float* ar,
                      const float* b, float* feat, float* out,
                      float* el, float* er, float* emax, float* denom, float* exbuf,
                      const int* src, const int* dst,
                      int Nn, int Din, int H, int E, hipStream_t stream) {
  const int Dout = H * HID;
  dim3 gg(Nn / 16, Dout / 64);
  gemm_wmma_f32<<<gg, 128, 0, stream>>>(h, W, feat, Nn, Din, Dout);

  const int NH = Nn * H;
  attn_prep<<<(NH + 255) / 256, 256, 0, stream>>>(feat, al, ar, el, er, emax, denom, NH, H);

  const int total = Nn * Dout;
  init_bias_k<<<(total + 255) / 256, 256, 0, stream>>>(out, b, total, Dout);

  const int Etot = E + Nn;
  const int EtotH = Etot * H;
  edge_max_k<<<(EtotH + 255) / 256, 256, 0, stream>>>(src, dst, E, EtotH, el, er, emax, H);
  edge_expsum_k<<<(EtotH + 255) / 256, 256, 0, stream>>>(src, dst, E, EtotH, el, er, emax,
                                                         exbuf, denom, H);
  edge_scatter_k<<<(Etot + 7) / 8, 256, 0, stream>>>(src, dst, E, Etot, feat, exbuf, denom,
                                                     out, H);
}

extern "C" void kernel_launch(void* const* d_in, const int* in_sizes, int n_in,
                              void* d_out, int out_size, void* d_ws, size_t ws_size,
                              hipStream_t stream) {
  const float* node = (const float*)d_in[0];
  const int* src = (const int*)d_in[1];
  const int* dst = (const int*)d_in[2];
  const float* W0 = (const float*)d_in[3];
  const float* al0 = (const float*)d_in[4];
  const float* ar0 = (const float*)d_in[5];
  const float* b0 = (const float*)d_in[6];
  const float* W1 = (const float*)d_in[7];
  const float* al1 = (const float*)d_in[8];
  const float* ar1 = (const float*)d_in[9];
  const float* b1 = (const float*)d_in[10];
  const float* W2 = (const float*)d_in[11];
  const float* al2 = (const float*)d_in[12];
  const float* ar2 = (const float*)d_in[13];
  const float* b2 = (const float*)d_in[14];

  const int IN = 128;
  const int Nn = in_sizes[0] / IN;   // 50000
  const int E = in_sizes[1];         // 400000
  const int Etot = E + Nn;

  // Workspace carve-up (floats):
  //   feat : N*256      (GEMM output of current layer)
  //   hbuf : N*256      (aggregated output of layer i == input of layer i+1)
  //   el/er/emax/denom : N*4 each
  //   exbuf : Etot*4
  float* ws = (float*)d_ws;
  const size_t NF = (size_t)Nn * 256;
  float* feat = ws;
  float* hbuf = feat + NF;
  float* el = hbuf + NF;
  float* er = el + (size_t)Nn * 4;
  float* emax = er + (size_t)Nn * 4;
  float* denom = emax + (size_t)Nn * 4;
  float* exbuf = denom + (size_t)Nn * 4;
  (void)ws_size; (void)n_in; (void)out_size; (void)Etot;

  // Layer 0: [N,128] -> 4 heads x 64 = [N,256]
  run_layer(node, W0, al0, ar0, b0, feat, hbuf, el, er, emax, denom, exbuf,
            src, dst, Nn, IN, 4, E, stream);
  // Layer 1: [N,256] -> [N,256]. GEMM fully consumes hbuf before init_bias
  // re-initializes it (stream-ordered), so hbuf doubles as the output.
  run_layer(hbuf, W1, al1, ar1, b1, feat, hbuf, el, er, emax, denom, exbuf,
            src, dst, Nn, 256, 4, E, stream);
  // Layer 2: [N,256] -> 1 head x 64 = [N,64], accumulate directly into d_out.
  run_layer(hbuf, W2, al2, ar2, b2, feat, (float*)d_out, el, er, emax, denom, exbuf,
            src, dst, Nn, 256, 1, E, stream);
}